// BidirectionalBlock_29781303230602
// MI455X (gfx1250) — compile-verified
//
#include <hip/hip_runtime.h>

// ---------------------------------------------------------------------------
// ChebConv (K=5) x 2 directions, N=50000, E=800000, C=64 -> out [N,128] f32.
// Sparse lhat passes are L2-resident atomic scatters; dense projections use
// CDNA5 V_WMMA_F32_16X16X4_F32 (full fp32 precision, wave32 layouts per ISA).
// ---------------------------------------------------------------------------

typedef float v2f __attribute__((ext_vector_type(2)));
typedef float v8f __attribute__((ext_vector_type(8)));

static __device__ __forceinline__ size_t row64(int n) { return (size_t)n * 64; }

// ---- utility kernels -------------------------------------------------------

__global__ __launch_bounds__(256) void k_zero(float* __restrict__ p, int n) {
    int i = blockIdx.x * 256 + threadIdx.x;
    if (i < n) p[i] = 0.0f;
}

// deg1[row] += ew ; deg2[col] += ew   (both directions in one pass)
__global__ __launch_bounds__(256) void k_deg(const float* __restrict__ ew,
                                             const int* __restrict__ row,
                                             const int* __restrict__ col,
                                             float* __restrict__ deg1,
                                             float* __restrict__ deg2, int E) {
    int e = blockIdx.x * 256 + threadIdx.x;
    if (e >= E) return;
    float w = ew[e];
    atomicAdd(&deg1[row[e]], w);
    atomicAdd(&deg2[col[e]], w);
}

// deg -> deg^{-1/2} in place (0 where deg<=0)
__global__ __launch_bounds__(256) void k_invsqrt(float* __restrict__ d, int n) {
    int i = blockIdx.x * 256 + threadIdx.x;
    if (i < n) {
        float v = d[i];
        d[i] = (v > 0.0f) ? rsqrtf(v) : 0.0f;
    }
}

// off_w = -scale * dis[row] * ew * dis[col] for each direction
__global__ __launch_bounds__(256) void k_edgew(const float* __restrict__ ew,
                                               const int* __restrict__ row,
                                               const int* __restrict__ col,
                                               const float* __restrict__ dis1,
                                               const float* __restrict__ dis2,
                                               float* __restrict__ ow1,
                                               float* __restrict__ ow2, int E) {
    int e = blockIdx.x * 256 + threadIdx.x;
    if (e >= E) return;
    const float scale = 2.0f / 3.0f;  // 2 / lambda_max
    int r = row[e], c = col[e];
    float w = ew[e];
    ow1[e] = -scale * dis1[r] * w * dis1[c];
    ow2[e] = -scale * dis2[r] * w * dis2[c];  // symmetric product, same either order
}

// out[n,0:64] = b1 ; out[n,64:128] = b2
__global__ __launch_bounds__(256) void k_bias(const float* __restrict__ b1,
                                              const float* __restrict__ b2,
                                              float* __restrict__ out, int n128) {
    int i = blockIdx.x * 256 + threadIdx.x;
    if (i >= n128) return;
    int j = i & 127;
    out[i] = (j < 64) ? b1[j] : b2[j - 64];
}

// dst = alpha_diag * v + beta * prev    (elementwise over N*64)
__global__ __launch_bounds__(256) void k_cheb_scale(const float* __restrict__ v,
                                                    const float* __restrict__ prev,
                                                    float* __restrict__ dst,
                                                    float alpha_diag, float beta,
                                                    int n64) {
    int i = blockIdx.x * 256 + threadIdx.x;
    if (i < n64) dst[i] = alpha_diag * v[i] + beta * prev[i];
}

// dst_feat[dstIdx[e], :] += alpha * ow[e] * v[srcIdx[e], :]
// one wave per edge: 32 lanes x 2 channels (float2 gather, 2 f32 atomics)
__global__ __launch_bounds__(256) void k_scatter(const float* __restrict__ v,
                                                 float* __restrict__ outf,
                                                 const int* __restrict__ srcIdx,
                                                 const int* __restrict__ dstIdx,
                                                 const float* __restrict__ ow,
                                                 float alpha, int E) {
    int e = blockIdx.x * 8 + (threadIdx.x >> 5);
    if (e >= E) return;
    int c2 = (threadIdx.x & 31) * 2;
    int s = srcIdx[e];
    int d = dstIdx[e];
    float w = alpha * ow[e];
    float2 val = *(const float2*)(v + row64(s) + c2);
    float* o = outf + row64(d) + c2;
    atomicAdd(o + 0, w * val.x);
    atomicAdd(o + 1, w * val.y);
}

// ---- WMMA fp32 GEMM accumulator -------------------------------------------
// out[n, colofs + j] += sum_c Tx[n,c] * W[c,j]   (Tx: [N,64], W: [64,64] row-major)
// Block: 256 threads = 8 waves; block tile = 128 rows x 64 cols.
// Wave w owns a 16-row strip and all 4 col tiles (4 x v8f accumulators):
// per k-step, one A fragment feeds 4 back-to-back WMMAs -> 64 WMMA / kernel.
//
// ISA VGPR layouts (wave32):
//   A (16x4):  lane L holds M=L&15, K = 2*(L>>4) + {0,1} in a[0],a[1]
//   B (4x16):  lane L holds N=L&15, K = 2*(L>>4) + {0,1} in b[0],b[1]
//   C (16x16): VGPR i, lane L -> row = i + 8*(L>>4), col = L&15
__global__ __launch_bounds__(256) void k_gemm_acc(const float* __restrict__ Tx,
                                                  const float* __restrict__ W,
                                                  float* __restrict__ out,
                                                  int colofs, int nRows) {
    __shared__ float Wl[64 * 64];    // B: [K=64][N=64], 16 KB
    __shared__ float Al[128 * 64];   // A: 128 staged Tx rows, 32 KB

    const int tid = threadIdx.x;
    const int row0b = blockIdx.x * 128;

    // stage W: 1024 float4, fully coalesced
    {
        const float4* W4 = (const float4*)W;
        float4* Wl4 = (float4*)Wl;
#pragma unroll
        for (int i = 0; i < 4; ++i) Wl4[tid + 256 * i] = W4[tid + 256 * i];
    }
    // stage A tile: 2048 float4, zero-fill OOB rows (guard outside hot loop)
    {
        float4* Al4 = (float4*)Al;
#pragma unroll
        for (int i = 0; i < 8; ++i) {
            int idx = tid + 256 * i;          // vec4 index; row = idx>>4
            int r = row0b + (idx >> 4);
            float4 v = make_float4(0.f, 0.f, 0.f, 0.f);
            if (r < nRows) v = ((const float4*)(Tx + row64(r)))[idx & 15];
            Al4[idx] = v;
        }
    }
    __syncthreads();

    const int wave = tid >> 5;           // 0..7 -> row tile within block
    const int lane = tid & 31;
    const int lm = lane & 15;
    const int lg = lane >> 4;

    const int rowt = wave * 16;          // row-tile base within Al
    const int row0 = row0b + rowt;       // global row base for this wave

    // C accumulators (4 col tiles); branch-free clamped loads
    v8f c0, c1, c2, c3;
    const int nclamp = nRows - 1;
#pragma unroll
    for (int i = 0; i < 8; ++i) {
        int r = row0 + i + lg * 8;
        bool ok = (r < nRows);
        int rc = ok ? r : nclamp;
        const float* orow = out + (size_t)rc * 128 + colofs + lm;
        float v0 = orow[0], v1 = orow[16], v2 = orow[32], v3 = orow[48];
        c0[i] = ok ? v0 : 0.f;
        c1[i] = ok ? v1 : 0.f;
        c2[i] = ok ? v2 : 0.f;
        c3[i] = ok ? v3 : 0.f;
    }

    const float* arow_l = Al + (rowt + lm) * 64;   // this lane's A row in LDS
    const float* bcol_l = Wl + lm;                 // this lane's B column base

#pragma unroll
    for (int kk = 0; kk < 16; ++kk) {
        const int kbase = kk * 4 + lg * 2;         // K of fragment element 0
        v2f a = *(const v2f*)(arow_l + kbase);     // ds_load_b64
        v2f b0, b1v, b2, b3;
        b0[0]  = bcol_l[(kbase + 0) * 64 + 0];
        b0[1]  = bcol_l[(kbase + 1) * 64 + 0];
        b1v[0] = bcol_l[(kbase + 0) * 64 + 16];
        b1v[1] = bcol_l[(kbase + 1) * 64 + 16];
        b2[0]  = bcol_l[(kbase + 0) * 64 + 32];
        b2[1]  = bcol_l[(kbase + 1) * 64 + 32];
        b3[0]  = bcol_l[(kbase + 0) * 64 + 48];
        b3[1]  = bcol_l[(kbase + 1) * 64 + 48];
        // (neg_a, A, neg_b, B, c_mod, C, reuse_a, reuse_b)
        c0 = __builtin_amdgcn_wmma_f32_16x16x4_f32(false, a, false, b0,  (short)0, c0, false, false);
        c1 = __builtin_amdgcn_wmma_f32_16x16x4_f32(false, a, false, b1v, (short)0, c1, false, false);
        c2 = __builtin_amdgcn_wmma_f32_16x16x4_f32(false, a, false, b2,  (short)0, c2, false, false);
        c3 = __builtin_amdgcn_wmma_f32_16x16x4_f32(false, a, false, b3,  (short)0, c3, false, false);
    }

#pragma unroll
    for (int i = 0; i < 8; ++i) {
        int r = row0 + i + lg * 8;
        if (r < nRows) {
            float* orow = out + (size_t)r * 128 + colofs + lm;
            orow[0]  = c0[i];
            orow[16] = c1[i];
            orow[32] = c2[i];
            orow[48] = c3[i];
        }
    }
}

// ---------------------------------------------------------------------------

extern "C" void kernel_launch(void* const* d_in, const int* in_sizes, int n_in,
                              void* d_out, int out_size, void* d_ws, size_t ws_size,
                              hipStream_t stream) {
    const float* x  = (const float*)d_in[0];   // [N,64]
    const int*   ei = (const int*)d_in[1];     // [2,E]
    const float* ew = (const float*)d_in[2];   // [E]
    const float* W1 = (const float*)d_in[3];   // [5,64,64]
    const float* b1 = (const float*)d_in[4];   // [64]
    const float* W2 = (const float*)d_in[5];   // [5,64,64]
    const float* b2 = (const float*)d_in[6];   // [64]
    float* out = (float*)d_out;                // [N,128]

    const int N = in_sizes[0] / 64;
    const int E = in_sizes[2];
    const int* row = ei;        // edge_index[0]
    const int* col = ei + E;    // edge_index[1]

    // workspace partition (~45 MB total)
    float* ws   = (float*)d_ws;
    float* deg1 = ws;                       // [N]  -> becomes dis1
    float* deg2 = deg1 + N;                 // [N]  -> becomes dis2
    float* ow1  = deg2 + N;                 // [E]
    float* ow2  = ow1 + E;                  // [E]
    float* tA   = ow2 + E;                  // [N,64]
    float* tB   = tA + (size_t)N * 64;      // [N,64]
    float* tC   = tB + (size_t)N * 64;      // [N,64]

    const int n64  = N * 64;
    const int n128 = N * 128;
    const float DIAG = 2.0f / 3.0f - 1.0f;  // scale - 1

    dim3 blk(256);
    int gN2   = (2 * N + 255) / 256;
    int gE    = (E + 255) / 256;
    int gE8   = (E + 7) / 8;
    int g64   = (n64 + 255) / 256;
    int g128  = (n128 + 255) / 256;
    int gGemm = (N + 127) / 128;

    // normalization
    k_zero<<<gN2, blk, 0, stream>>>(deg1, 2 * N);
    k_deg<<<gE, blk, 0, stream>>>(ew, row, col, deg1, deg2, E);
    k_invsqrt<<<gN2, blk, 0, stream>>>(deg1, 2 * N);
    k_edgew<<<gE, blk, 0, stream>>>(ew, row, col, deg1, deg2, ow1, ow2, E);

    // out = [b1 | b2] broadcast
    k_bias<<<g128, blk, 0, stream>>>(b1, b2, out, n128);

    for (int dir = 0; dir < 2; ++dir) {
        const int*   src  = (dir == 0) ? col : row;  // gather side
        const int*   dst  = (dir == 0) ? row : col;  // scatter side
        const float* ow   = (dir == 0) ? ow1 : ow2;
        const float* W    = (dir == 0) ? W1 : W2;
        const int    cofs = (dir == 0) ? 0 : 64;

        // k = 0 : out += x @ W[0]
        k_gemm_acc<<<gGemm, blk, 0, stream>>>(x, W, out, cofs, N);

        // k = 1 : tA = lhat(x) = DIAG*x + scatter(ow * x[src])
        k_cheb_scale<<<g64, blk, 0, stream>>>(x, x, tA, DIAG, 0.0f, n64);
        k_scatter<<<gE8, blk, 0, stream>>>(x, tA, src, dst, ow, 1.0f, E);
        k_gemm_acc<<<gGemm, blk, 0, stream>>>(tA, W + 1 * 4096, out, cofs, N);

        // k = 2 : tB = 2*lhat(tA) - x
        k_cheb_scale<<<g64, blk, 0, stream>>>(tA, x, tB, 2.0f * DIAG, -1.0f, n64);
        k_scatter<<<gE8, blk, 0, stream>>>(tA, tB, src, dst, ow, 2.0f, E);
        k_gemm_acc<<<gGemm, blk, 0, stream>>>(tB, W + 2 * 4096, out, cofs, N);

        // k = 3 : tC = 2*lhat(tB) - tA
        k_cheb_scale<<<g64, blk, 0, stream>>>(tB, tA, tC, 2.0f * DIAG, -1.0f, n64);
        k_scatter<<<gE8, blk, 0, stream>>>(tB, tC, src, dst, ow, 2.0f, E);
        k_gemm_acc<<<gGemm, blk, 0, stream>>>(tC, W + 3 * 4096, out, cofs, N);

        // k = 4 : tA = 2*lhat(tC) - tB   (tA's old contents no longer needed)
        k_cheb_scale<<<g64, blk, 0, stream>>>(tC, tB, tA, 2.0f * DIAG, -1.0f, n64);
        k_scatter<<<gE8, blk, 0, stream>>>(tC, tA, src, dst, ow, 2.0f, E);
        k_gemm_acc<<<gGemm, blk, 0, stream>>>(tA, W + 4 * 4096, out, cofs, N);
    }
}